// MF_54193897341080
// MI455X (gfx1250) — compile-verified
//
#include <hip/hip_runtime.h>

// Matrix-factorization prediction + loss for MI455X (gfx1250, wave32).
//
// pred[b,k] = dot(user_weight[user[b]] + user_bias[user[b]],
//                 item_weight[item[b,k]] + item_bias[item[b,k]]) + bias
// loss = mean((pred-target)^2) + 0.01*mean(||ue||) + 0.01*mean(||ie||)
//
// Memory-bound (~210MB of item-row gathers -> ~9us at 23.3 TB/s).
// Dot products run on V_WMMA_F32_16X16X4_F32: one wave = one batch row b,
// one 16-item k-tile; A = 16x4 item-embedding tile, B = 4x16 with all
// columns equal to the user-embedding chunk, D diagonal = predictions.
// Invalid k-lanes (K=50 padded to 64) clamp to item row 0 and are masked
// only at the outputs (pred store / mse / norm), never inside the WMMA
// pipeline -- EXEC stays all-ones and no per-chunk cndmasks are needed.

typedef float v2f __attribute__((ext_vector_type(2)));
typedef float v8f __attribute__((ext_vector_type(8)));

#define NUM_B 16384
#define NUM_K 50
#define NUM_H 64
#define REG_C 0.01f

__global__ __launch_bounds__(128) void mf_pred_kernel(
    const float* __restrict__ user_weight,
    const float* __restrict__ item_weight,
    const float* __restrict__ user_bias,
    const float* __restrict__ item_bias,
    const float* __restrict__ bias,
    const float* __restrict__ target,
    const int*   __restrict__ user,
    const int*   __restrict__ item,
    float*       __restrict__ pred_out,  // d_out: [B*K] preds (+ loss slot at end)
    float*       __restrict__ ws)        // [0,B): mse, [B,2B): ie-norm, [2B,3B): ue-norm
{
    const int b    = blockIdx.x;
    const int tid  = threadIdx.x;
    const int wave = tid >> 5;     // 4 waves, one 16-item k-tile each
    const int lane = tid & 31;
    const int m    = lane & 15;    // item-in-tile / output column
    const int half = lane >> 4;    // h-pair selector within each K=4 chunk
    const int k0   = wave * 16;

    // Wave-uniform user-side values (compiler scalarizes these to SGPRs).
    const int   u  = user[b];
    const float ub = user_bias[u];
    const float gb = bias[0];
    const float* __restrict__ uerow = user_weight + (size_t)u * NUM_H;

    // Per-lane item-side values (lanes m and m+16 share item m of the tile).
    const int  k     = k0 + m;
    const bool valid = (k < NUM_K);
    const int  idx   = valid ? item[(size_t)b * NUM_K + k] : 0;  // clamp keeps loads in-bounds
    const float ibv  = item_bias[idx];
    const float* __restrict__ iwrow = item_weight + (size_t)idx * NUM_H;

    v8f   acc = {};     // 16x16 f32 C/D accumulator
    float isq = 0.0f;   // per-lane half-row sum of squares of ie (for ||ie||)

    // H=64 as 16 chunks of K=4. A-lane(m, half) needs item floats {4c+2*half, +1};
    // B-slot(v, half) needs user float ue[4c + 2*half + v]: same float2 pattern.
    #pragma unroll
    for (int c = 0; c < 16; ++c) {
        const int h = c * 4 + half * 2;

        const v2f av = *(const v2f*)(iwrow + h);
        const float a0 = av.x + ibv;
        const float a1 = av.y + ibv;

        const v2f uv = *(const v2f*)(uerow + h);
        const float b0 = uv.x + ub;
        const float b1 = uv.y + ub;

        const v2f A  = {a0, a1};
        const v2f Bm = {b0, b1};
        // (neg_a, A, neg_b, B, c_mod, C, reuse_a, reuse_b)
        acc = __builtin_amdgcn_wmma_f32_16x16x4_f32(false, A, false, Bm,
                                                    (short)0, acc, false, false);
        isq += a0 * a0 + a1 * a1;
    }

    // Diagonal extraction: lane holds column m, rows r+8*half in acc[r].
    // Diagonal (row==col) lives in lanes 0..7 (r=m) and 24..31 (r=m-8).
    float msq = 0.0f;
    #pragma unroll
    for (int r = 0; r < 8; ++r) {
        const int M = r + half * 8;
        if (m == M) {
            const int kk = k0 + M;
            if (kk < NUM_K) {
                const float p = acc[r] + gb;
                pred_out[(size_t)b * NUM_K + kk] = p;
                const float d = p - target[(size_t)b * NUM_K + kk];
                msq = d * d;
            }
        }
    }

    // ||ie|| : combine the two h-halves of each item row across the lane pair.
    const float isq_full = isq + __shfl_xor(isq, 16, 32);
    float inorm = 0.0f;
    if (half == 0 && valid) inorm = sqrtf(isq_full);

    // Wave reduction (fixed order -> deterministic).
    float wm = msq, wi = inorm;
    #pragma unroll
    for (int off = 16; off > 0; off >>= 1) {
        wm += __shfl_xor(wm, off, 32);
        wi += __shfl_xor(wi, off, 32);
    }

    // ||ue|| once per block (wave 0): 64 floats = float2 per lane.
    float unorm = 0.0f;
    if (wave == 0) {
        const v2f uv = *(const v2f*)(uerow + lane * 2);
        const float e0 = uv.x + ub, e1 = uv.y + ub;
        float us = e0 * e0 + e1 * e1;
        #pragma unroll
        for (int off = 16; off > 0; off >>= 1) us += __shfl_xor(us, off, 32);
        unorm = sqrtf(us);
    }

    __shared__ float s_m[4], s_i[4], s_u;
    if (lane == 0) {
        s_m[wave] = wm;
        s_i[wave] = wi;
        if (wave == 0) s_u = unorm;
    }
    __syncthreads();
    if (tid == 0) {
        ws[b]             = s_m[0] + s_m[1] + s_m[2] + s_m[3];
        ws[NUM_B + b]     = s_i[0] + s_i[1] + s_i[2] + s_i[3];
        ws[2 * NUM_B + b] = s_u;
    }
}

__global__ __launch_bounds__(256) void mf_loss_kernel(const float* __restrict__ ws,
                                                      float* __restrict__ out)
{
    __shared__ float sm[256], si[256], su[256];
    const int t = threadIdx.x;
    float am = 0.0f, ai = 0.0f, au = 0.0f;
    for (int i = t; i < NUM_B; i += 256) {
        am += ws[i];
        ai += ws[NUM_B + i];
        au += ws[2 * NUM_B + i];
    }
    sm[t] = am; si[t] = ai; su[t] = au;
    __syncthreads();
    #pragma unroll
    for (int s = 128; s > 0; s >>= 1) {
        if (t < s) { sm[t] += sm[t + s]; si[t] += si[t + s]; su[t] += su[t + s]; }
        __syncthreads();
    }
    if (t == 0) {
        const float inv_bk = 1.0f / (float)(NUM_B * NUM_K);
        const float mse = sm[0] * inv_bk;
        const float reg = REG_C * (su[0] / (float)NUM_B) + REG_C * (si[0] * inv_bk);
        out[(size_t)NUM_B * NUM_K] = mse + reg;
    }
}

extern "C" void kernel_launch(void* const* d_in, const int* in_sizes, int n_in,
                              void* d_out, int out_size, void* d_ws, size_t ws_size,
                              hipStream_t stream) {
    const float* user_weight = (const float*)d_in[0];
    const float* item_weight = (const float*)d_in[1];
    const float* user_bias   = (const float*)d_in[2];
    const float* item_bias   = (const float*)d_in[3];
    const float* bias        = (const float*)d_in[4];
    const float* target      = (const float*)d_in[5];
    const int*   user        = (const int*)d_in[6];
    const int*   item        = (const int*)d_in[7];
    float* out = (float*)d_out;
    float* ws  = (float*)d_ws;

    mf_pred_kernel<<<NUM_B, 128, 0, stream>>>(user_weight, item_weight, user_bias,
                                              item_bias, bias, target, user, item,
                                              out, ws);
    mf_loss_kernel<<<1, 256, 0, stream>>>(ws, out);
}